// KBpA_49692771615174
// MI455X (gfx1250) — compile-verified
//
#include <hip/hip_runtime.h>

#define HH 512
#define WW 512
#define KSZ 21
#define PADR 10
#define NPIX (HH * WW)
#define NCEN 1024
#define TILE_W 36   // 16 + 2*10 halo

typedef __attribute__((ext_vector_type(2))) float v2f;
typedef __attribute__((ext_vector_type(8))) float v8f;

// -log2(e) / (2 * SDP2) with SDP2 = 4.0  ->  exp(-big/8) == exp2(s*big)
#define SSCALE (-0.18033688011112042f)

// ---------------------------------------------------------------------------
// Kernel 1: 21x21 same-padding cross-correlation on both coord channels,
// then pack per-pixel WMMA A-rows: (dp_y, dp_x, |dp|^2, 1.0)
// ---------------------------------------------------------------------------
__global__ __launch_bounds__(256) void conv_pack_kernel(
    const float* __restrict__ betas,   // [NPIX,2] interleaved
    const float* __restrict__ kern,    // [21*21]
    float4* __restrict__ Apack)        // [NPIX]
{
    __shared__ float s0[TILE_W * TILE_W];
    __shared__ float s1[TILE_W * TILE_W];
    __shared__ float sk[KSZ * KSZ];

    const int tid = threadIdx.y * 16 + threadIdx.x;
    const int bx = blockIdx.x * 16, by = blockIdx.y * 16;

    for (int i = tid; i < TILE_W * TILE_W; i += 256) {
        int gy = by - PADR + i / TILE_W;
        int gx = bx - PADR + i % TILE_W;
        float a = 0.f, b = 0.f;
        if (gy >= 0 && gy < HH && gx >= 0 && gx < WW) {
            int n = gy * WW + gx;
            a = betas[2 * n];
            b = betas[2 * n + 1];
        }
        s0[i] = a;
        s1[i] = b;
    }
    for (int i = tid; i < KSZ * KSZ; i += 256) sk[i] = kern[i];
    __syncthreads();

    const int ty = threadIdx.y, tx = threadIdx.x;
    float acc0 = 0.f, acc1 = 0.f;
    for (int ky = 0; ky < KSZ; ++ky) {
        const float* r0 = &s0[(ty + ky) * TILE_W + tx];
        const float* r1 = &s1[(ty + ky) * TILE_W + tx];
        const float* kr = &sk[ky * KSZ];
#pragma unroll
        for (int kx = 0; kx < KSZ; ++kx) {
            float kv = kr[kx];
            acc0 = __builtin_fmaf(r0[kx], kv, acc0);
            acc1 = __builtin_fmaf(r1[kx], kv, acc1);
        }
    }
    const int y = by + ty, x = bx + tx;
    const float dpy = (float)y - acc0;
    const float dpx = (float)x - acc1;
    const float p2 = dpy * dpy + dpx * dpx;
    Apack[y * WW + x] = make_float4(dpy, dpx, p2, 1.0f);
}

// ---------------------------------------------------------------------------
// Kernel 2: pack per-center WMMA B-columns: (-2s*cy, -2s*cx, s, s*|c|^2)
// so that A.B = s * (p2 + c2 - 2 dp.c) = log2-arg of the Gaussian directly.
// ---------------------------------------------------------------------------
__global__ void center_pack_kernel(const float* __restrict__ centers,  // [NCEN,2]
                                   float4* __restrict__ Bpack)         // [NCEN]
{
    const int m = blockIdx.x * blockDim.x + threadIdx.x;
    if (m < NCEN) {
        const float cy = centers[2 * m];
        const float cx = centers[2 * m + 1];
        const float c2 = cy * cy + cx * cx;
        Bpack[m] = make_float4(-2.0f * SSCALE * cy, -2.0f * SSCALE * cx,
                               SSCALE, SSCALE * c2);
    }
}

// ---------------------------------------------------------------------------
// Kernel 3: one wave per 16-pixel tile. 64x v_wmma_f32_16x16x4_f32 over all
// centers, v_exp_f32 per element, fold GEMV with alphas into per-lane FMAs,
// 4-step shfl_xor butterfly (within lane groups of 16) to finish the row sums.
// ---------------------------------------------------------------------------
__global__ __launch_bounds__(256) void rbf_wmma_kernel(
    const float4* __restrict__ Bpack,   // [NCEN]
    const float* __restrict__ alphas,   // [NCEN]
    const float* __restrict__ Apack,    // [NPIX*4] floats
    float* __restrict__ out)            // [NPIX]
{
    __shared__ float sB[NCEN * 4];   // 16 KB: packed B columns
    __shared__ float sAl[NCEN];      //  4 KB: alphas

    const int tid = threadIdx.x;
    for (int i = tid; i < NCEN; i += 256) {
        float4 b = Bpack[i];
        ((float4*)sB)[i] = b;
        sAl[i] = alphas[i];
    }
    __syncthreads();

    const int lane = tid & 31;
    const int wave = tid >> 5;
    const int tile = blockIdx.x * 8 + wave;   // 16384 tiles total
    const int p0 = tile * 16;

    // A fragment: lane l -> pixel row (l%16), K components 2*(l/16), 2*(l/16)+1
    const int half = lane >> 4;
    const int sub  = lane & 15;
    const v2f a = *(const v2f*)(Apack + (size_t)(p0 + sub) * 4 + half * 2);

    float acc[8] = {0.f, 0.f, 0.f, 0.f, 0.f, 0.f, 0.f, 0.f};

    for (int t = 0; t < NCEN / 16; ++t) {
        // B fragment: lane l -> center col (l%16), K components 2*(l/16)..
        const v2f b = *(const v2f*)(&sB[t * 64 + sub * 4 + half * 2]);
        v8f c = {};
        c = __builtin_amdgcn_wmma_f32_16x16x4_f32(
                /*neg_a=*/false, a, /*neg_b=*/false, b,
                /*c_mod=*/(short)0, c, /*reuse_a=*/false, /*reuse_b=*/false);
        const float al = sAl[t * 16 + sub];
#pragma unroll
        for (int r = 0; r < 8; ++r)
            acc[r] = __builtin_fmaf(__builtin_amdgcn_exp2f(c[r]), al, acc[r]);
    }

    // Reduce across the 16 center-lanes; lanes 0-15 and 16-31 stay separate.
#pragma unroll
    for (int r = 0; r < 8; ++r) {
        float v = acc[r];
        v += __shfl_xor(v, 1, 32);
        v += __shfl_xor(v, 2, 32);
        v += __shfl_xor(v, 4, 32);
        v += __shfl_xor(v, 8, 32);
        acc[r] = v;
    }

    // VGPR r holds row r (lanes<16) or row r+8 (lanes>=16)
    if (sub == 0) {
        const int base = p0 + half * 8;
#pragma unroll
        for (int r = 0; r < 8; ++r) out[base + r] = acc[r];
    }
}

// ---------------------------------------------------------------------------
extern "C" void kernel_launch(void* const* d_in, const int* in_sizes, int n_in,
                              void* d_out, int out_size, void* d_ws, size_t ws_size,
                              hipStream_t stream) {
    (void)in_sizes; (void)n_in; (void)out_size; (void)ws_size;
    const float* betas   = (const float*)d_in[0];   // [NPIX,2]
    const float* alphas  = (const float*)d_in[1];   // [NCEN,1]
    const float* kern    = (const float*)d_in[2];   // [21,21]
    // d_in[3] all_pixels unused: pixel coords are the implicit (y,x) grid
    const float* centers = (const float*)d_in[4];   // [NCEN,2]
    float* out = (float*)d_out;

    float4* Apack = (float4*)d_ws;                                  // 4 MB
    float4* Bpack = (float4*)((char*)d_ws + (size_t)NPIX * 16);     // 16 KB

    conv_pack_kernel<<<dim3(WW / 16, HH / 16), dim3(16, 16), 0, stream>>>(
        betas, kern, Apack);
    center_pack_kernel<<<NCEN / 256, 256, 0, stream>>>(centers, Bpack);
    rbf_wmma_kernel<<<NPIX / 16 / 8, 256, 0, stream>>>(
        Bpack, alphas, (const float*)Apack, out);
}